// EvolvingFilteredResonance_22016002360007
// MI455X (gfx1250) — compile-verified
//
#include <hip/hip_runtime.h>
#include <hip/hip_bf16.h>
#include <math.h>

// ---------------------------------------------------------------------------
// EvolvingFilteredResonance for MI455X (gfx1250)
//   out = [ start_resonance (16,64,32768),
//           end_resonance   (16,64,32768),
//           crossfade_stack (16,64,32768,2) ]  (float32, flat-concatenated)
//
// Strategy:
//   * 1 workgroup (256 threads = 8 wave32) per (b,e) row; full row lives in
//     LDS as 32768 complex f32 = 256KB (CDNA5: 320KB LDS per workgroup).
//   * forward in-place mixed-radix DIF [16,16,16,8]  -> digit-reversed bins
//   * pointwise: Z = X * (f_start + i * f_end)   (both filters, one pass)
//   * inverse in-place mixed-radix DIT [8,16,16,16] -> natural order
//     Re(z) = start output, Im(z) = end output   (two real irffts in one)
//   * radix-16 butterflies = 16x16 complex DFT matmuls on v_wmma_f32_16x16x4_f32
//   * input row DMA'd into LDS with global_load_async_to_lds (ASYNCcnt path)
//   * outputs streamed with non-temporal stores (write-once, 512MB total)
// ---------------------------------------------------------------------------

#define N_FFT   32768
#define ROWS    1024          // BATCH * N_EVENTS
#define TPB     256
#define PI_F    3.14159265358979323846f

#ifndef __has_builtin
#define __has_builtin(x) 0
#endif

#if __has_builtin(__builtin_amdgcn_global_load_async_to_lds_b32) && \
    __has_builtin(__builtin_amdgcn_s_wait_asynccnt)
#define USE_ASYNC_LDS 1
#else
#define USE_ASYNC_LDS 0
#endif

typedef float v8f __attribute__((ext_vector_type(8)));
typedef float v2f __attribute__((ext_vector_type(2)));

typedef __attribute__((address_space(1))) int as1_int;
typedef __attribute__((address_space(3))) int as3_int;

// D(16x16) += A(16x4) * B(4x16), fp32 WMMA (CDNA5 wave32)
__device__ __forceinline__ v8f wmma_f32_16x16x4(v2f a, v2f b, v8f c) {
    return __builtin_amdgcn_wmma_f32_16x16x4_f32(
        /*neg_a=*/false, a, /*neg_b=*/false, b,
        /*c_mod=*/(short)0, c, /*reuse_a=*/false, /*reuse_b=*/false);
}

__device__ __forceinline__ float2 cmul(float2 a, float2 b) {
    return make_float2(a.x * b.x - a.y * b.y, a.x * b.y + a.y * b.x);
}
__device__ __forceinline__ float2 cadd(float2 a, float2 b) {
    return make_float2(a.x + b.x, a.y + b.y);
}
__device__ __forceinline__ float2 csub(float2 a, float2 b) {
    return make_float2(a.x - b.x, a.y - b.y);
}

// ---------------------------------------------------------------------------
// Radix-16 stage (in place), span m (power of two), transform length 16*m.
// DIR = -1: forward DIF (twiddle W_n^{-u*j} applied AFTER the DFT)
// DIR = +1: inverse DIT (twiddle W_n^{+j*r} applied BEFORE the DFT)
// The 16-point DFT itself is D = DFT16 * Xgroups done with WMMA:
//   A operand = DFT16 rows (M=u, K=r), B operand = 16 gathered groups (K=r, N=g)
//   complex matmul via 2 accumulators, 16 wmma ops per 16-group chunk.
// 2048 groups / stage -> 128 chunks -> 16 chunks per wave (8 waves).
// ---------------------------------------------------------------------------
template <int DIR>
__device__ void radix16_stage(float2* __restrict__ X, int m, int mlog2,
                              int wave, int lane) {
    const int   hi    = lane >> 4;       // 0 or 1 (lane half)
    const int   M     = lane & 15;       // A-matrix row carried by this lane
    const int   g     = lane & 15;       // B/D column (group) for this lane
    const int   nlog2 = mlog2 + 4;       // log2 of transform length
    const int   nmask = (1 << nlog2) - 1;
    const float invn  = 1.0f / (float)(1 << nlog2);

    // Constant DFT16 matrix, A-operand layout:
    //   lanes 0-15 : M=lane,    VGPR0 -> K=4c+0, VGPR1 -> K=4c+1
    //   lanes 16-31: M=lane-16, VGPR0 -> K=4c+2, VGPR1 -> K=4c+3
    v2f dr[4], di[4], ndi[4];
#pragma unroll
    for (int c = 0; c < 4; ++c) {
        int   k0 = 4 * c + 2 * hi;
        float a0 = (2.0f * PI_F / 16.0f) * (float)((M * k0) & 15);
        float a1 = (2.0f * PI_F / 16.0f) * (float)((M * (k0 + 1)) & 15);
        float s0, c0, s1, c1;
        __sincosf(a0, &s0, &c0);
        __sincosf(a1, &s1, &c1);
        v2f t;
        t.x = c0;                t.y = c1;                dr[c]  = t;
        t.x = (float)DIR * s0;   t.y = (float)DIR * s1;   di[c]  = t;
        t.x = -(float)DIR * s0;  t.y = -(float)DIR * s1;  ndi[c] = t;
    }

#pragma unroll 1
    for (int t16 = 0; t16 < 16; ++t16) {
        const int CH   = wave * 16 + t16;          // chunk id [0,128)
        const int G    = CH * 16 + g;              // this lane's group id
        const int j    = G & (m - 1);
        const int blk  = G >> mlog2;
        const int base = (blk << nlog2) + j;

        // Gather B operand: K = r = 4c + v + 2*hi, N = g
        v2f br[4], bi[4];
#pragma unroll
        for (int c = 0; c < 4; ++c) {
            const int r0 = 4 * c + 2 * hi;
            float2 x0 = X[base + (r0 << mlog2)];
            float2 x1 = X[base + ((r0 + 1) << mlog2)];
            if (DIR > 0) {  // inverse DIT: input twiddle W_n^{+j*r}
                float s, cv;
                float a0 = 2.0f * PI_F * (float)((j * r0) & nmask) * invn;
                float a1 = 2.0f * PI_F * (float)((j * (r0 + 1)) & nmask) * invn;
                __sincosf(a0, &s, &cv);  x0 = cmul(x0, make_float2(cv, s));
                __sincosf(a1, &s, &cv);  x1 = cmul(x1, make_float2(cv, s));
            }
            v2f t;
            t.x = x0.x; t.y = x1.x; br[c] = t;
            t.x = x0.y; t.y = x1.y; bi[c] = t;
        }

        // Complex matmul:  Yr = Dr*Xr + (-Di)*Xi ;  Yi = Dr*Xi + Di*Xr
        v8f accr = {0.f, 0.f, 0.f, 0.f, 0.f, 0.f, 0.f, 0.f};
        v8f acci = {0.f, 0.f, 0.f, 0.f, 0.f, 0.f, 0.f, 0.f};
#pragma unroll
        for (int c = 0; c < 4; ++c) {
            accr = wmma_f32_16x16x4(dr[c], br[c], accr);
            acci = wmma_f32_16x16x4(dr[c], bi[c], acci);
        }
#pragma unroll
        for (int c = 0; c < 4; ++c) {
            accr = wmma_f32_16x16x4(ndi[c], bi[c], accr);
            acci = wmma_f32_16x16x4(di[c],  br[c], acci);
        }

        // Scatter D: VGPR v, lanes 0-15 -> row u=v, lanes 16-31 -> row u=v+8
#pragma unroll
        for (int v = 0; v < 8; ++v) {
            const int u = v + 8 * hi;
            float2 y = make_float2(accr[v], acci[v]);
            if (DIR < 0) {  // forward DIF: output twiddle W_n^{-u*j}
                float s, cv;
                float a = -2.0f * PI_F * (float)((u * j) & nmask) * invn;
                __sincosf(a, &s, &cv);
                y = cmul(y, make_float2(cv, s));
            }
            X[base + (u << mlog2)] = y;
        }
    }
}

// ---------------------------------------------------------------------------
// Radix-8 stage with span m=1 (contiguous groups of 8), natural-order 8-pt DFT
// done as 3 register radix-2 layers + bit-reversed scatter. 4096 groups,
// 16 per thread. DIR=-1 forward, +1 inverse.
// ---------------------------------------------------------------------------
template <int DIR>
__device__ void radix8_stage(float2* __restrict__ X, int tid) {
    const float  C  = 0.70710678118654752f;
    const float2 w1 = make_float2(C, (float)DIR * C);   // W8^1
    const float2 w2 = make_float2(0.0f, (float)DIR);    // W8^2 = W4^1
    const float2 w3 = make_float2(-C, (float)DIR * C);  // W8^3
#pragma unroll 1
    for (int q = 0; q < 16; ++q) {
        const int Gbase = 8 * (tid + TPB * q);
        float2 x0 = X[Gbase + 0], x1 = X[Gbase + 1];
        float2 x2 = X[Gbase + 2], x3 = X[Gbase + 3];
        float2 x4 = X[Gbase + 4], x5 = X[Gbase + 5];
        float2 x6 = X[Gbase + 6], x7 = X[Gbase + 7];
        // layer 1 (m=4)
        float2 a0 = cadd(x0, x4), a1 = cadd(x1, x5);
        float2 a2 = cadd(x2, x6), a3 = cadd(x3, x7);
        float2 b0 = csub(x0, x4);
        float2 b1 = cmul(csub(x1, x5), w1);
        float2 b2 = cmul(csub(x2, x6), w2);
        float2 b3 = cmul(csub(x3, x7), w3);
        // layer 2 (m=2)
        float2 c0 = cadd(a0, a2), c2 = csub(a0, a2);
        float2 c1 = cadd(a1, a3), c3 = cmul(csub(a1, a3), w2);
        float2 d0 = cadd(b0, b2), d2 = csub(b0, b2);
        float2 d1 = cadd(b1, b3), d3 = cmul(csub(b1, b3), w2);
        // layer 3 (m=1) + bitrev3 scatter {0,4,2,6,1,5,3,7}
        X[Gbase + 0] = cadd(c0, c1);  X[Gbase + 4] = csub(c0, c1);
        X[Gbase + 2] = cadd(c2, c3);  X[Gbase + 6] = csub(c2, c3);
        X[Gbase + 1] = cadd(d0, d1);  X[Gbase + 5] = csub(d0, d1);
        X[Gbase + 3] = cadd(d2, d3);  X[Gbase + 7] = csub(d2, d3);
    }
}

// ---------------------------------------------------------------------------
// Main FFT bandpass kernel: one row per workgroup.
// ---------------------------------------------------------------------------
__global__ __launch_bounds__(TPB)
void fft_bandpass_kernel(const float* __restrict__ res,
                         const float* __restrict__ sm, const float* __restrict__ ss,
                         const float* __restrict__ em, const float* __restrict__ es,
                         float* __restrict__ out) {
    extern __shared__ float2 X[];   // 32768 complex f32 = 256 KB
    const int rid  = blockIdx.x;
    const int tid  = threadIdx.x;
    const int lane = tid & 31;
    const int wave = tid >> 5;

    const float* xin = res + (size_t)rid * N_FFT;

#if USE_ASYNC_LDS
    // Zero the imaginary dwords with plain DS stores (disjoint addresses from
    // the async-DMA'd real dwords, so no ordering requirement), then DMA the
    // real samples straight into the interleaved LDS slots (ASYNCcnt-tracked).
    {
        float* Xs = (float*)X;
        for (int i = tid; i < N_FFT; i += TPB) Xs[2 * i + 1] = 0.0f;
        as1_int* gsrc = (as1_int*)xin;   // C-style cast: generic->AS1, drops const
        as3_int* ldst = (as3_int*)X;     // C-style cast: generic->AS3
        for (int i = tid; i < N_FFT; i += TPB) {
            __builtin_amdgcn_global_load_async_to_lds_b32(
                gsrc + i,        // global dword i
                ldst + 2 * i,    // interleaved real slot (byte offset 8*i)
                /*offset=*/0, /*cpol=*/0);
        }
        __builtin_amdgcn_s_wait_asynccnt(0);
    }
#else
    for (int i = tid; i < N_FFT; i += TPB)
        X[i] = make_float2(__builtin_nontemporal_load(xin + i), 0.0f);
#endif
    __syncthreads();

    // Forward DIF, radices [16,16,16,8] -> digit-reversed spectrum
    radix16_stage<-1>(X, 2048, 11, wave, lane);  __syncthreads();
    radix16_stage<-1>(X,  128,  7, wave, lane);  __syncthreads();
    radix16_stage<-1>(X,    8,  3, wave, lane);  __syncthreads();
    radix8_stage<-1>(X, tid);                    __syncthreads();

    // Pointwise: Z[k] = X[k] * (f_start(k) + i * f_end(k))
    // bin k recovered from digit-reversed position p.
    const float m1  = sm[rid], m2 = em[rid];
    const float sd1 = ss[rid], sd2 = es[rid];
    const float is1 = -0.5f / (sd1 * sd1 + 1e-8f);
    const float is2 = -0.5f / (sd2 * sd2 + 1e-8f);
#pragma unroll 1
    for (int q = 0; q < N_FFT / TPB; ++q) {
        const int p  = tid + TPB * q;
        const int k1 = p >> 11, k2 = (p >> 7) & 15, k3 = (p >> 3) & 15, k4 = p & 7;
        const int k  = k1 | (k2 << 4) | (k3 << 8) | (k4 << 12);
        const int km = N_FFT - k;
        const int kk = (k < km) ? k : km;              // mirrored bin (Hermitian)
        const float f   = (float)kk * (1.0f / 16384.0f);
        const float df1 = f - m1, df2 = f - m2;
        const float f1  = __expf(df1 * df1 * is1);
        const float f2  = __expf(df2 * df2 * is2);
        const float2 z  = X[p];
        X[p] = make_float2(z.x * f1 - z.y * f2, z.x * f2 + z.y * f1);
    }
    __syncthreads();

    // Inverse DIT, radices [8,16,16,16] -> natural order (unnormalized)
    radix8_stage<+1>(X, tid);                    __syncthreads();
    radix16_stage<+1>(X,    8,  3, wave, lane);  __syncthreads();
    radix16_stage<+1>(X,  128,  7, wave, lane);  __syncthreads();
    radix16_stage<+1>(X, 2048, 11, wave, lane);  __syncthreads();

    // Re -> start_resonance, Im -> end_resonance (scale 1/N).
    // Write-once streams: non-temporal stores keep them out of L2.
    const float inv = 1.0f / (float)N_FFT;
    float* o1 = out + (size_t)rid * N_FFT;
    float* o2 = out + (size_t)ROWS * N_FFT + (size_t)rid * N_FFT;
    for (int i = tid; i < N_FFT; i += TPB) {
        const float2 z = X[i];
        __builtin_nontemporal_store(z.x * inv, o1 + i);
        __builtin_nontemporal_store(z.y * inv, o2 + i);
    }
}

// ---------------------------------------------------------------------------
// Crossfade kernel: env[f] = decay^(f+1) on 128 frames, linear-upsample to
// 32768 (align_corners=False semantics), write (cf, 1-cf) as float2.
// ---------------------------------------------------------------------------
__global__ __launch_bounds__(TPB)
void crossfade_kernel(const float* __restrict__ decays, float* __restrict__ out) {
    const int rid = blockIdx.x;
    const float d    = decays[rid];
    const float sig  = 1.0f / (1.0f + __expf(-d));
    const float dec  = 0.02f + sig * (0.98f * 0.95f);   // base + sig*(1-base)*0.95
    const float ld   = __logf(dec + 1e-12f);
    v2f* o = (v2f*)out + (size_t)rid * N_FFT;
    for (int i = threadIdx.x; i < N_FFT; i += TPB) {
        float coord = ((float)i + 0.5f) * (1.0f / 256.0f) - 0.5f;  // scale = 128/32768
        coord = fminf(fmaxf(coord, 0.0f), 127.0f);
        const float lo = floorf(coord);
        const float w  = coord - lo;
        const int  ilo = (int)lo;
        const int  ihi = (ilo + 1 < 127) ? (ilo + 1) : 127;
        const float e0 = __expf((float)(ilo + 1) * ld);
        const float e1 = __expf((float)(ihi + 1) * ld);
        const float cf = e0 * (1.0f - w) + e1 * w;
        v2f val; val.x = cf; val.y = 1.0f - cf;
        __builtin_nontemporal_store(val, o + i);
    }
}

// ---------------------------------------------------------------------------
extern "C" void kernel_launch(void* const* d_in, const int* in_sizes, int n_in,
                              void* d_out, int out_size, void* d_ws, size_t ws_size,
                              hipStream_t stream) {
    const float* res = (const float*)d_in[0];   // (16,64,32768)
    const float* dec = (const float*)d_in[1];   // (16,64,1)
    const float* sm  = (const float*)d_in[2];   // start means
    const float* ss  = (const float*)d_in[3];   // start stds
    const float* em  = (const float*)d_in[4];   // end means
    const float* es  = (const float*)d_in[5];   // end stds
    float* out = (float*)d_out;

    (void)in_sizes; (void)n_in; (void)out_size; (void)d_ws; (void)ws_size;

    fft_bandpass_kernel<<<ROWS, TPB, N_FFT * sizeof(float2), stream>>>(
        res, sm, ss, em, es, out);
    crossfade_kernel<<<ROWS, TPB, 0, stream>>>(
        dec, out + 2 * (size_t)ROWS * N_FFT);
}